// OurButterflyLayer_82660940579338
// MI455X (gfx1250) — compile-verified
//
#include <hip/hip_runtime.h>

// Butterfly layer for MI455X (gfx1250).
//
// Roofline: 32MB in + 128MB out ~= 160MB  ->  ~7us at 23.3 TB/s HBM.
// Compute after factoring B^T into 3 radix-16 block-diagonal passes is only
// ~3.2 GFLOP, done with V_WMMA_F32_16X16X4_F32 (16x16x16 product = 4 chained
// K=4 WMMAs). All intermediate state lives in LDS (16 rows x 4096 f32
// ~= 257KB of the 320KB WGP budget); HBM is touched once per element.

#define LOGN_      12
#define NN         4096
#define IN_SIZE_   1024
#define OUT_SIZE_  4096
#define BATCH_     8192
#define TILE_ROWS  16
#define ROW_STRIDE (NN + 16)   // floats, padded row pitch in LDS
#define THREADS_   256

typedef __attribute__((ext_vector_type(2))) float v2f;
typedef __attribute__((ext_vector_type(8))) float v8f;

// 4-bit XOR swizzle on the low nibble of a column index: keeps the stride-16
// (pass 1) and stride-256 (pass 2) LDS gathers spread across the 64 banks.
__device__ __forceinline__ int sw(int j) {
  return j ^ (((j >> 4) ^ (j >> 8)) & 15);
}

// ---------------------------------------------------------------------------
// Kernel 1: fold each pass's 4 butterfly stages into dense 16x16 block
// matrices M (768 blocks total: 3 passes x 256 blocks), stored pre-swizzled
// in the WMMA B-matrix lane layout:
//   B[k][n] = M[n][k];  lane L (n = L&15, hi = L>>4), wmma step w:
//   float2 = ( M[n][4w+2hi], M[n][4w+2hi+1] )
// so the main kernel's B-operand loads are coalesced float2 reads.
// 16 lanes cooperate per block; row mixing via __shfl_xor (wave32).
// ---------------------------------------------------------------------------
__global__ __launch_bounds__(THREADS_)
void butterfly_factors(const float* __restrict__ P, float2* __restrict__ G) {
  const int gid     = blockIdx.x * THREADS_ + threadIdx.x;  // 48*256 = 12288
  const int t       = gid & 15;        // row of the 16x16 block matrix
  const int blockId = gid >> 4;        // 0..767
  const int pass    = blockId >> 8;    // 0..2
  const int B       = blockId & 255;
  const int S       = 1 << (4 * pass);               // stride 16^pass
  const int base    = (B / S) * (S * 16) + (B % S);  // hi*16^{p+1} + lo

  float M[16];
#pragma unroll
  for (int c = 0; c < 16; ++c) M[c] = (c == t) ? 1.0f : 0.0f;

#pragma unroll
  for (int q = 0; q < 4; ++q) {
    const int i  = 4 * pass + q;       // global stage index
    const int m  = 1 << q;             // xor mask on the radix-16 digit
    const int jr = base + t * S;
    const int jp = base + (t ^ m) * S;
    const float a = P[(2 * i)     * NN + jr];   // a_i[j]
    const float b = P[(2 * i + 1) * NN + jp];   // b_i[j ^ 2^i]
#pragma unroll
    for (int c = 0; c < 16; ++c) {
      const float oldv = M[c];
      const float part = __shfl_xor(oldv, m, 32);  // M[t^m][c] (pre-update)
      M[c] = a * oldv + b * part;
    }
  }

#pragma unroll
  for (int w = 0; w < 4; ++w)
#pragma unroll
    for (int hi = 0; hi < 2; ++hi) {
      const float2 v = make_float2(M[4 * w + 2 * hi], M[4 * w + 2 * hi + 1]);
      G[((size_t)blockId * 4 + w) * 32 + (t + 16 * hi)] = v;
    }
}

// ---------------------------------------------------------------------------
// Kernel 2: per workgroup, 16 batch rows live in LDS (16 x 4096 f32).
// 3 in-place passes; pass p applies Y_tile(16x16) = X_tile(16x16) @ Mblk^T
// for each of its 256 disjoint column blocks (pass 0: only the 64 blocks
// covering the nonzero first 1024 columns) via 4 chained
// V_WMMA_F32_16X16X4_F32. Then gather through out_row_indices and store.
// ---------------------------------------------------------------------------
__global__ __launch_bounds__(THREADS_)
void butterfly_apply(const float* __restrict__ inp,
                     const float2* __restrict__ G,
                     const int* __restrict__ idx,
                     float* __restrict__ out) {
  extern __shared__ float tile[];                 // [TILE_ROWS][ROW_STRIDE]
  const int  tid     = threadIdx.x;
  const int  lane    = tid & 31;                  // wave32
  // readfirstlane => compiler sees a wave-uniform (SGPR) value, so the block
  // loop below is scalar: no exec-mask loop machinery, EXEC provably all-1s
  // around the WMMAs (ISA requirement).
  const int  wave    = __builtin_amdgcn_readfirstlane(tid >> 5);  // 0..7
  const long rowBase = (long)blockIdx.x * TILE_ROWS;

  // ---- stage in the 16x1024 input tile (swizzled), zero-fill cols 1024..4095
  for (int i = tid; i < TILE_ROWS * (IN_SIZE_ / 4); i += THREADS_) {
    const int r  = i >> 8;                        // IN_SIZE_/4 == 256
    const int c4 = i & 255;
    const float4 v = ((const float4*)inp)[(rowBase + r) * (IN_SIZE_ / 4) + c4];
    const int c = c4 * 4;
    float* trow = &tile[r * ROW_STRIDE];
    trow[sw(c + 0)] = v.x;
    trow[sw(c + 1)] = v.y;
    trow[sw(c + 2)] = v.z;
    trow[sw(c + 3)] = v.w;
  }
  for (int i = tid; i < TILE_ROWS * (NN - IN_SIZE_); i += THREADS_) {
    const int r = i >> 10;                        // (NN-IN_SIZE_)/3 ... use shifts:
    // i in [0, 16*3072): r = i / 3072, c = 1024 + i % 3072 — rewrite with
    // a 4096-based split to stay on shifts: i2 = i + (i/3072)*1024 is messy;
    // keep the simple form (compiler strength-reduces the div/mod):
    const int rr = i / (NN - IN_SIZE_);
    const int c  = IN_SIZE_ + (i - rr * (NN - IN_SIZE_));
    (void)r;
    tile[rr * ROW_STRIDE + sw(c)] = 0.0f;
  }
  __syncthreads();

  const int m  = lane & 15;   // A: batch row / D: output column digit
  const int hi = lane >> 4;   // half-wave selects K pair (A/B) or row half (D)

#pragma unroll
  for (int pass = 0; pass < 3; ++pass) {
    const int S    = 1 << (4 * pass);
    const int nblk = (pass == 0) ? (IN_SIZE_ / 16) : 256;  // skip all-zero blocks
    const float2* Gp = G + (size_t)pass * 256 * 4 * 32;

    // Software-pipelined scalar loop: G operands for block B+8 are fetched
    // while block B's LDS reads + WMMA chain execute, hiding global latency.
    float2 g[4];
#pragma unroll
    for (int w = 0; w < 4; ++w)
      g[w] = Gp[((size_t)wave * 4 + w) * 32 + lane];

    for (int B = wave; B < nblk; B += 8) {
      const int Bn = B + 8;
      float2 gn[4];
      if (Bn < nblk) {
#pragma unroll
        for (int w = 0; w < 4; ++w)
          gn[w] = Gp[((size_t)Bn * 4 + w) * 32 + lane];
      }

      const int base = (B / S) * (S * 16) + (B % S);
      v8f acc = {};
#pragma unroll
      for (int w = 0; w < 4; ++w) {
        const int kk = 4 * w + 2 * hi;            // K digit pair for this half-wave
        v2f a;                                    // A 16x4 f32 VGPR layout
        a.x = tile[m * ROW_STRIDE + sw(base + kk * S)];
        a.y = tile[m * ROW_STRIDE + sw(base + (kk + 1) * S)];
        v2f b; b.x = g[w].x; b.y = g[w].y;        // B 4x16 f32 (pre-swizzled)
        acc = __builtin_amdgcn_wmma_f32_16x16x4_f32(
            /*neg_a=*/false, a, /*neg_b=*/false, b,
            /*c_mod=*/(short)0, acc, /*reuse_a=*/false, /*reuse_b=*/false);
      }
      // D 16x16 f32: lane holds col N = m, rows M = v + 8*hi. In-place store:
      // this wave's blocks own disjoint column sets, so no intra-pass barrier.
#pragma unroll
      for (int v = 0; v < 8; ++v)
        tile[(v + 8 * hi) * ROW_STRIDE + sw(base + m * S)] = acc[v];

#pragma unroll
      for (int w = 0; w < 4; ++w) g[w] = gn[w];
    }
    __syncthreads();
  }

  // ---- gather out_row_indices once into registers, then write coalesced
  int srccol[OUT_SIZE_ / THREADS_];               // 16 column indices per thread
#pragma unroll
  for (int t = 0; t < OUT_SIZE_ / THREADS_; ++t)
    srccol[t] = sw(idx[tid + THREADS_ * t]);

  for (int r = 0; r < TILE_ROWS; ++r) {
    const float* trow = &tile[r * ROW_STRIDE];
    float* orow = &out[(rowBase + r) * OUT_SIZE_];
#pragma unroll
    for (int t = 0; t < OUT_SIZE_ / THREADS_; ++t)
      orow[tid + THREADS_ * t] = trow[srccol[t]];
  }
}

extern "C" void kernel_launch(void* const* d_in, const int* in_sizes, int n_in,
                              void* d_out, int out_size, void* d_ws, size_t ws_size,
                              hipStream_t stream) {
  const float* inp = (const float*)d_in[0];  // [8192][1024] f32
  const float* P   = (const float*)d_in[1];  // [24][4096]   f32
  const int*   idx = (const int*)d_in[2];    // [4096]       i32
  float*       out = (float*)d_out;          // [8192][4096] f32
  float2*      G   = (float2*)d_ws;          // 768 blocks * 128 float2 = 786,432 B

  butterfly_factors<<<48, THREADS_, 0, stream>>>(P, G);

  const size_t shmem = (size_t)TILE_ROWS * ROW_STRIDE * sizeof(float);  // 263,168 B
  butterfly_apply<<<BATCH_ / TILE_ROWS, THREADS_, shmem, stream>>>(inp, G, idx, out);
}